// MeshSamplingCSR_62345745269208
// MI455X (gfx1250) — compile-verified
//
#include <hip/hip_runtime.h>
#include <stdint.h>

// Problem constants (from reference)
#define B_ 8
#define N_ 50000
#define C_ 128
#define M_ 25000
#define K_ 3

#define WAVES_PER_BLOCK 8
#define ROWS_PER_WAVE   4            // M_ % ROWS_PER_WAVE == 0 -> no b crossing
#define THREADS         (WAVES_PER_BLOCK * 32)

#define ROW_BYTES       (C_ * 4)             // 512 B per x row
#define STAGE_BYTES     (K_ * ROW_BYTES)     // 1536 B per pipeline stage
#define WAVE_LDS_BYTES  (2 * STAGE_BYTES)    // double buffer
#define BLOCK_LDS_BYTES (WAVES_PER_BLOCK * WAVE_LDS_BYTES)   // 24576 B

typedef float v4f __attribute__((ext_vector_type(4)));
typedef __attribute__((address_space(3))) const v4f* lds_cv4f;

__global__ __launch_bounds__(THREADS) void mesh_sampling_kernel(
    const float* __restrict__ x,     // [B, N, C]
    const int*   __restrict__ cols,  // [M, K]
    const float* __restrict__ vals,  // [M, K]
    float*       __restrict__ out,   // [B, M, C]
    int totalRows)                   // B*M
{
  const int lane = (int)(threadIdx.x & 31u);
  const int wv   = (int)(threadIdx.x >> 5);
  // Everything per-wave is made scalar up front.
  const int gw      = __builtin_amdgcn_readfirstlane((int)blockIdx.x * WAVES_PER_BLOCK + wv);
  const int rowBase = gw * ROWS_PER_WAVE;
  if (rowBase >= totalRows) return;

  const int b  = rowBase / M_;            // scalar magic-mul, SALU
  const int m0 = rowBase - b * M_;        // rows m0..m0+3 all in batch b

  const uint64_t xb      = (uint64_t)(uintptr_t)x + (uint64_t)b * ((uint64_t)N_ * ROW_BYTES);
  const uint32_t voff    = (uint32_t)lane * 16u;                    // per-lane global offset
  const uint32_t ldsLane = (uint32_t)wv * WAVE_LDS_BYTES + voff;    // per-lane LDS base

  const int*   cb = cols + (size_t)m0 * K_;   // uniform -> SMEM loads
  const float* vb = vals + (size_t)m0 * K_;   // uniform -> SMEM loads

  // Issue one x-row per instruction: 32 lanes x 16 B = 512 B, GVS (saddr) form.
  auto issue = [&](int j) {
    const uint32_t lb = ldsLane + (uint32_t)(j & 1) * STAGE_BYTES;
#pragma unroll
    for (int k = 0; k < K_; ++k) {
      const int ck = cb[j * K_ + k];                                   // scalar load
      const uint64_t sa = xb + (uint64_t)(uint32_t)ck * (uint64_t)ROW_BYTES;
      const uint32_t la = lb + (uint32_t)(k * ROW_BYTES);
      asm volatile("global_load_async_to_lds_b128 %0, %1, %2"
                   :: "v"(la), "v"(voff), "s"(sa) : "memory");
    }
  };

  // Combine staged rows with wave-uniform weights; streaming NT store.
  auto compute = [&](int j) {
    const float w0 = vb[j * K_ + 0];
    const float w1 = vb[j * K_ + 1];
    const float w2 = vb[j * K_ + 2];
    lds_cv4f s = (lds_cv4f)(ldsLane + (uint32_t)(j & 1) * STAGE_BYTES);
    v4f g0 = s[0];      // ds_load_b128, offset 0
    v4f g1 = s[32];     // ds_load_b128, offset 512
    v4f g2 = s[64];     // ds_load_b128, offset 1024
    v4f r = g0 * w0 + g1 * w1 + g2 * w2;
    v4f* op = (v4f*)(out + (size_t)(rowBase + j) * C_) + lane;
    __builtin_nontemporal_store(r, op);   // keep per-batch x slice resident in L2
  };

  // Software pipeline: prefetch row j+1 while combining row j.
  issue(0);
#pragma unroll
  for (int j = 0; j < ROWS_PER_WAVE; ++j) {
    if (j + 1 < ROWS_PER_WAVE) {
      issue(j + 1);
      // 6 async ops in flight; async loads complete in order, so <=3 means
      // stage j's three row-transfers have landed in LDS.
      asm volatile("s_wait_asynccnt 3" ::: "memory");
    } else {
      asm volatile("s_wait_asynccnt 0" ::: "memory");
    }
    compute(j);
  }
}

extern "C" void kernel_launch(void* const* d_in, const int* in_sizes, int n_in,
                              void* d_out, int out_size, void* d_ws, size_t ws_size,
                              hipStream_t stream) {
  (void)in_sizes; (void)n_in; (void)d_ws; (void)ws_size; (void)out_size;
  const float* x    = (const float*)d_in[0];
  const int*   cols = (const int*)d_in[1];
  const float* vals = (const float*)d_in[2];
  float* out = (float*)d_out;

  const int totalRows = B_ * M_;                                            // 200000
  const int numWaves  = (totalRows + ROWS_PER_WAVE - 1) / ROWS_PER_WAVE;    // 50000
  const int grid      = (numWaves + WAVES_PER_BLOCK - 1) / WAVES_PER_BLOCK; // 6250

  // Dynamic LDS: staging buffers live at raw offsets 0..BLOCK_LDS_BYTES-1,
  // matching both the async-write side and the addrspace(3) read side.
  mesh_sampling_kernel<<<grid, THREADS, BLOCK_LDS_BYTES, stream>>>(
      x, cols, vals, out, totalRows);
}